// PManifoldLayer_996432413004
// MI455X (gfx1250) — compile-verified
//
#include <hip/hip_runtime.h>
#include <math.h>

#define EPSF 1e-7f
#define KK 32
#define MM 256
#define NN 4096

typedef __attribute__((ext_vector_type(2))) float v2f;
typedef __attribute__((ext_vector_type(8))) float v8f;

// ---------------------------------------------------------------------------
// Kernel 1: per-point norms, class weights, theta norms, zero per_k buffer.
// ---------------------------------------------------------------------------
__global__ void prep_kernel(const float* __restrict__ dgm,
                            const int* __restrict__ hom,
                            const float* __restrict__ theta,
                            const float* __restrict__ class_w,
                            float* __restrict__ dn2,
                            float* __restrict__ wn,
                            float* __restrict__ tk2,
                            float* __restrict__ per_k) {
  int i = blockIdx.x * blockDim.x + threadIdx.x;  // 0..4095
  const float* row = dgm + (size_t)i * MM;
  float s = 0.0f;
  for (int m = 0; m < MM; ++m) s = fmaf(row[m], row[m], s);
  dn2[i] = s;
  wn[i] = class_w[hom[i]];
  if (i < KK) {
    const float* tr = theta + (size_t)i * MM;
    float t = 0.0f;
    for (int m = 0; m < MM; ++m) t = fmaf(tr[m], tr[m], t);
    tk2[i] = t;
  }
  // zero per_k (32*256 = 8192 floats, 4096 threads -> 2 each)
  per_k[i] = 0.0f;
  per_k[i + NN] = 0.0f;
}

// ---------------------------------------------------------------------------
// Kernel 2: GEMM1  dot[k,n] = theta[k,:] . dgm[n,:]  via V_WMMA_F32_16X16X4_F32
// fused epilogue: zn2 -> Poincare log-map scale -> a[k,n] = w[n]*c[k,n]
// 512 tiles of 16x16, one wave per tile, 8 waves/block, 64 blocks.
// ---------------------------------------------------------------------------
__global__ void gemm1_kernel(const float* __restrict__ theta,
                             const float* __restrict__ dgm,
                             const float* __restrict__ dn2,
                             const float* __restrict__ tk2,
                             const float* __restrict__ wn,
                             float* __restrict__ a_buf) {
  const int wave = threadIdx.x >> 5;
  const int lane = threadIdx.x & 31;
  const int tile = blockIdx.x * 8 + wave;   // 0..511
  const int kt = tile >> 8;                 // 0..1   (k tile)
  const int nt = tile & 255;                // 0..255 (n tile)
  const int half = lane >> 4;               // 0..1
  const int low  = lane & 15;               // 0..15

  v8f acc = {};
  // A frag: theta row (kt*16+low), K index = mb + 2*half + j
  const float* arow = theta + (size_t)(kt * 16 + low) * MM + 2 * half;
  // B frag: B[m_red, n] = dgm[n, m_red]; n = nt*16+low
  const float* brow = dgm + (size_t)(nt * 16 + low) * MM + 2 * half;

  for (int mb = 0; mb < MM; mb += 4) {
    v2f a = *(const v2f*)(arow + mb);
    v2f b = *(const v2f*)(brow + mb);
    acc = __builtin_amdgcn_wmma_f32_16x16x4_f32(false, a, false, b,
                                                (short)0, acc, false, false);
  }

  // Epilogue: C layout lane=(half,low), vgpr v -> (M = v + 8*half, N = low)
  const int n = nt * 16 + low;
  const float dn2v = dn2[n];
  const float wv = wn[n];
#pragma unroll
  for (int v = 0; v < 8; ++v) {
    const int k = kt * 16 + v + 8 * half;
    float zn2 = dn2v + 2.0f * acc[v] + tk2[k];
    float s1 = 1.0f / (1.0f + sqrtf(1.0f + zn2));  // x = s1 * z
    float nx = sqrtf(fmaxf(zn2, 0.0f)) * s1;       // ||x||
    nx = fminf(fmaxf(nx, EPSF), 1.0f - EPSF);
    float at = 0.5f * logf((1.0f + nx) / (1.0f - nx));  // artanh
    float c = (at / nx) * s1;                      // t = c * z
    a_buf[(size_t)k * NN + n] = wv * c;
  }
}

// ---------------------------------------------------------------------------
// Kernel 3: GEMM2  per_k[k,m] += sum_n a[k,n]*dgm[n,m]
// 32 tiles of 16x16, each split across 8 waves over the N=4096 reduction,
// fp32 atomic accumulation. 32 blocks x 256 threads.
// ---------------------------------------------------------------------------
__global__ void gemm2_kernel(const float* __restrict__ a_buf,
                             const float* __restrict__ dgm,
                             float* __restrict__ per_k) {
  const int wave = threadIdx.x >> 5;   // reduction split 0..7
  const int lane = threadIdx.x & 31;
  const int tile = blockIdx.x;         // 0..31
  const int kt = tile >> 4;            // 0..1
  const int mt = tile & 15;            // 0..15
  const int half = lane >> 4;
  const int low  = lane & 15;

  v8f acc = {};
  // A frag: a_buf row (kt*16+low), K(reduction) index = nb + 2*half + j
  const float* arow = a_buf + (size_t)(kt * 16 + low) * NN + 2 * half;
  const int n0 = wave * 512;
  const int mcol = mt * 16 + low;

  for (int nb = n0; nb < n0 + 512; nb += 4) {
    v2f a = *(const v2f*)(arow + nb);
    const int nr = nb + 2 * half;
    v2f b;
    b.x = dgm[(size_t)nr * MM + mcol];
    b.y = dgm[(size_t)(nr + 1) * MM + mcol];
    acc = __builtin_amdgcn_wmma_f32_16x16x4_f32(false, a, false, b,
                                                (short)0, acc, false, false);
  }

#pragma unroll
  for (int v = 0; v < 8; ++v) {
    const int k = kt * 16 + v + 8 * half;
    atomicAdd(&per_k[k * MM + mcol], acc[v]);
  }
}

// ---------------------------------------------------------------------------
// Kernel 4: finalize. S[k] = sum_n a[k,n]; cumsum over k of
// (per_k + theta*S[k]); then y = chart(exp_map_0(cum)) =
// 2*tanh(n)/n * cum / max(1 - tanh(n)^2, eps). One block of 256 threads.
// ---------------------------------------------------------------------------
__global__ void finalize_kernel(const float* __restrict__ per_k,
                                const float* __restrict__ theta,
                                const float* __restrict__ a_buf,
                                float* __restrict__ out) {
  __shared__ float red[256];
  __shared__ float Ssh[KK];
  const int t = threadIdx.x;  // m index 0..255

  // --- S[k]: 8 partials per k ---
  {
    const int k = t >> 3;
    const int part = t & 7;
    const float* ar = a_buf + (size_t)k * NN + part * 512;
    float s = 0.0f;
    for (int i = 0; i < 512; ++i) s += ar[i];
    red[t] = s;
    __syncthreads();
    if (part == 0) {
      float tot = 0.0f;
      for (int j = 0; j < 8; ++j) tot += red[t + j];
      Ssh[k] = tot;
    }
    __syncthreads();
  }

  float cum = 0.0f;
  for (int k = 0; k < KK; ++k) {
    cum += per_k[k * MM + t] + theta[k * MM + t] * Ssh[k];
    // block reduction of ||cum||^2
    red[t] = cum * cum;
    __syncthreads();
    for (int off = 128; off > 0; off >>= 1) {
      if (t < off) red[t] += red[t + off];
      __syncthreads();
    }
    const float nrm2 = red[0];
    __syncthreads();  // protect red[] before next iteration's write
    const float nn = fmaxf(sqrtf(nrm2), EPSF);
    const float th = tanhf(nn);
    const float denom = fmaxf(1.0f - th * th, EPSF);
    out[k * MM + t] = 2.0f * (th / nn) * cum / denom;
  }
}

// ---------------------------------------------------------------------------
extern "C" void kernel_launch(void* const* d_in, const int* in_sizes, int n_in,
                              void* d_out, int out_size, void* d_ws, size_t ws_size,
                              hipStream_t stream) {
  (void)in_sizes; (void)n_in; (void)out_size; (void)ws_size;
  const float* dgm    = (const float*)d_in[0];
  const int*   hom    = (const int*)d_in[1];
  const float* theta  = (const float*)d_in[2];
  const float* clsw   = (const float*)d_in[3];

  float* ws    = (float*)d_ws;
  float* a_buf = ws;                 // 32*4096 = 131072 floats
  float* per_k = ws + 131072;        // 32*256  =   8192 floats
  float* dn2   = ws + 139264;        // 4096
  float* wn    = ws + 143360;        // 4096
  float* tk2   = ws + 147456;        // 32

  prep_kernel<<<16, 256, 0, stream>>>(dgm, hom, theta, clsw, dn2, wn, tk2, per_k);
  gemm1_kernel<<<64, 256, 0, stream>>>(theta, dgm, dn2, tk2, wn, a_buf);
  gemm2_kernel<<<32, 256, 0, stream>>>(a_buf, dgm, per_k);
  finalize_kernel<<<1, 256, 0, stream>>>(per_k, theta, a_buf, (float*)d_out);
}